// FFTConv1d_thresh_36215164240736
// MI455X (gfx1250) — compile-verified
//
#include <hip/hip_runtime.h>
#include <hip/hip_bf16.h>

#define FFT_L   4096
#define FFT_NT  256
#define PI_F    3.14159265358979323846f

// ---------------------------------------------------------------------------
// Kernel 1: per-(b,c) spectral denoise.
//   load 62x62 image, zero-pad to 64x64 (pad (1,1)), flatten to L=4096,
//   forward radix-2 DIF FFT (natural -> bit-reversed),
//   threshold: zero bins where |Re| < 0.5 (elementwise, order-independent),
//   inverse radix-2 DIT FFT (bit-reversed -> natural), scale 1/L,
//   write real part to workspace x_t[b,c,64,64].
// Entirely LDS-resident: 32 KB per workgroup (well under 320 KB/WGP).
// ---------------------------------------------------------------------------
__global__ __launch_bounds__(FFT_NT) void fft_denoise_kernel(
    const float* __restrict__ x, float* __restrict__ xt) {
  __shared__ float re[FFT_L];
  __shared__ float im[FFT_L];
  const int img = blockIdx.x;  // b*32 + c, 128 total
  const int tid = threadIdx.x;
  const float* xi = x + (size_t)img * 62 * 62;

  // Load with ZeroPad2d((1,1,1,1)).
  for (int idx = tid; idx < FFT_L; idx += FFT_NT) {
    int i = idx >> 6, j = idx & 63;
    float v = 0.0f;
    if (i >= 1 && i <= 62 && j >= 1 && j <= 62) v = xi[(i - 1) * 62 + (j - 1)];
    re[idx] = v;
    im[idx] = 0.0f;
  }
  __syncthreads();

  // Forward DIF: spans 2048 -> 1, twiddle after subtract.
  for (int s = 11; s >= 0; --s) {
    const int half = 1 << s;
    const float w = -PI_F / (float)half;
    for (int t = tid; t < FFT_L / 2; t += FFT_NT) {
      int pos = t & (half - 1);
      int i0 = ((t >> s) << (s + 1)) + pos;
      int i1 = i0 + half;
      float ar = re[i0], ai = im[i0];
      float br = re[i1], bi = im[i1];
      float sr = ar - br, si = ai - bi;
      float sn, cs;
      __sincosf(w * (float)pos, &sn, &cs);
      re[i0] = ar + br;
      im[i0] = ai + bi;
      re[i1] = sr * cs - si * sn;
      im[i1] = sr * sn + si * cs;
    }
    __syncthreads();
  }

  // Threshold (spectrum is in bit-reversed storage order; elementwise op,
  // so the permutation is irrelevant).
  for (int idx = tid; idx < FFT_L; idx += FFT_NT) {
    if (fabsf(re[idx]) < 0.5f) {
      re[idx] = 0.0f;
      im[idx] = 0.0f;
    }
  }
  __syncthreads();

  // Inverse DIT: spans 1 -> 2048, conjugate twiddle before add/sub.
  for (int s = 0; s <= 11; ++s) {
    const int half = 1 << s;
    const float w = PI_F / (float)half;
    for (int t = tid; t < FFT_L / 2; t += FFT_NT) {
      int pos = t & (half - 1);
      int i0 = ((t >> s) << (s + 1)) + pos;
      int i1 = i0 + half;
      float sn, cs;
      __sincosf(w * (float)pos, &sn, &cs);
      float br = re[i1], bi = im[i1];
      float tr = br * cs - bi * sn;
      float ti = br * sn + bi * cs;
      float ar = re[i0], ai = im[i0];
      re[i0] = ar + tr;
      im[i0] = ai + ti;
      re[i1] = ar - tr;
      im[i1] = ai - ti;
    }
    __syncthreads();
  }

  float* xo = xt + (size_t)img * FFT_L;
  const float scale = 1.0f / (float)FFT_L;
  for (int idx = tid; idx < FFT_L; idx += FFT_NT) xo[idx] = re[idx] * scale;
}

// ---------------------------------------------------------------------------
// Kernel 2: 3x3 valid cross-correlation of x_t with W (+bias) as implicit
// GEMM on V_WMMA_F32_16X16X4_F32 (fp32-exact, matches reference dtype).
//   GEMM: D[M=64 out-ch][N=b*62*62 spatial], K = 32*9 = 288 (72 wmma steps).
// Layouts (ISA 7.12.2, wave32):
//   A 16x4 f32:  lane -> M = lane%16, K = vgpr + 2*(lane>=16)
//   B 4x16 f32:  lane -> N = lane%16, K = vgpr + 2*(lane>=16)
//   C/D 16x16:   lane -> N = lane%16, M = vgpr + 8*(lane>=16)
// One workgroup = 4 waves = all 64 out-channels for one 16-wide row chunk.
// All 4 waves share the same B tile, so the 32ch x 3row x 18col input patch
// is staged into LDS once per block (6.9 KB) instead of 4x72x256B of
// redundant scattered global loads (~10.7x traffic reduction incl. window
// reuse). A-side W loads have no intra-block reuse -> stay in global/L2,
// prefetched ahead of the LDS stage.
// Uniform control flow around WMMAs (EXEC all-1s); masked store at the end.
// ---------------------------------------------------------------------------
typedef __attribute__((ext_vector_type(2))) float v2f;
typedef __attribute__((ext_vector_type(8))) float v8f;

#define PATCH_ELEMS (32 * 3 * 18)  // 1728 floats = 6912 B

__global__ __launch_bounds__(128) void conv3x3_wmma_kernel(
    const float* __restrict__ xt, const float* __restrict__ Wf,
    const float* __restrict__ bias, float* __restrict__ out) {
  __shared__ float xs[PATCH_ELEMS];  // [c][kh][col] = [32][3][18]

  const int lane = threadIdx.x & 31;
  const int wave = threadIdx.x >> 5;  // 0..3
  const int o_base = wave * 16;       // M tile
  const int j0 = blockIdx.x * 16;     // 0,16,32,48 (62 cols -> last chunk masked)
  const int i = blockIdx.y;           // 0..61
  const int b = blockIdx.z;           // 0..3

  const int hi = lane >> 4;  // lane-half selects K pair {0,1} vs {2,3}
  const int n = lane & 15;   // A row (M) and B column (N)

  const float* wrow = Wf + (size_t)(o_base + n) * 288;  // W[o][c*9+kh*3+kw]
  const float* xb = xt + (size_t)b * 32 * FFT_L;

  // Start pulling this lane's W slab toward the caches while we stage LDS.
  __builtin_prefetch(wrow, 0, 0);

  // Cooperative patch stage: rows i..i+2, cols j0..j0+17, all 32 channels.
  // Column clamp to 63 keeps reads in-bounds; clamped columns only feed
  // outputs that the store mask discards (j >= 62).
  for (int e = threadIdx.x; e < PATCH_ELEMS; e += 128) {
    int c = e / 54;
    int rem = e - c * 54;
    int kh = rem / 18;
    int col = rem - kh * 18;
    int jc = j0 + col;
    if (jc > 63) jc = 63;
    xs[e] = xb[((size_t)c * 64 + (i + kh)) * 64 + jc];
  }
  __syncthreads();

  v8f acc = {};
  for (int kk = 0; kk < 288; kk += 4) {
    const int ka = kk + 2 * hi;

    v2f a;  // A[m][ka], A[m][ka+1] -- contiguous pair, merges to b64 load
    a.x = wrow[ka];
    a.y = wrow[ka + 1];

    // B rows ka, ka+1 at im2col column n, served from LDS.
    int c0 = ka / 9, r0 = ka - c0 * 9;
    int kh0 = r0 / 3, kw0 = r0 - kh0 * 3;
    int kb = ka + 1;
    int c1 = kb / 9, r1 = kb - c1 * 9;
    int kh1 = r1 / 3, kw1 = r1 - kh1 * 3;
    v2f bv;
    bv.x = xs[c0 * 54 + kh0 * 18 + n + kw0];
    bv.y = xs[c1 * 54 + kh1 * 18 + n + kw1];

    // 8 args: (neg_a, A, neg_b, B, c_mod, C, reuse_a, reuse_b)
    acc = __builtin_amdgcn_wmma_f32_16x16x4_f32(false, a, false, bv, (short)0,
                                                acc, false, false);
  }

  if (j0 + n < 62) {
#pragma unroll
    for (int r = 0; r < 8; ++r) {
      int o = o_base + r + 8 * hi;  // D: M = vgpr + 8*(lane>=16)
      float v = acc[r] + bias[o];
      out[(((size_t)b * 64 + o) * 62 + i) * 62 + (j0 + n)] = v;
    }
  }
}

// ---------------------------------------------------------------------------
// Launch: inputs (setup_inputs order): x[4,32,62,62] f32, W[64,32,3,3] f32,
// b[64] f32. Output: [4,64,62,62] f32. Workspace: x_t = 4*32*64*64 f32 = 2MB.
// ---------------------------------------------------------------------------
extern "C" void kernel_launch(void* const* d_in, const int* in_sizes, int n_in,
                              void* d_out, int out_size, void* d_ws,
                              size_t ws_size, hipStream_t stream) {
  const float* x = (const float*)d_in[0];
  const float* W = (const float*)d_in[1];
  const float* bias = (const float*)d_in[2];
  float* out = (float*)d_out;
  float* xt = (float*)d_ws;  // [4,32,64,64]

  fft_denoise_kernel<<<dim3(128), dim3(FFT_NT), 0, stream>>>(x, xt);
  conv3x3_wmma_kernel<<<dim3(4, 62, 4), dim3(128), 0, stream>>>(xt, W, bias,
                                                                out);
}